// ChebConv_54451595379259
// MI455X (gfx1250) — compile-verified
//
#include <hip/hip_runtime.h>
#include <hip/hip_bf16.h>
#include <stddef.h>

// Problem constants (from the reference)
#define V_N    50000
#define E_N    800000
#define B_N    4
#define CIN    64
#define COUT   128
#define KORD   4

typedef __attribute__((ext_vector_type(2))) float v2f;
typedef __attribute__((ext_vector_type(8))) float v8f;

// ---------------------------------------------------------------------------
// Kernel 1: CSR row pointers from sorted COO rows (binary search, per vertex)
// ---------------------------------------------------------------------------
__global__ void cheb_rowptr_kernel(const int* __restrict__ rows, int* __restrict__ rp) {
    int v = blockIdx.x * blockDim.x + threadIdx.x;
    if (v > V_N) return;
    int lo = 0, hi = E_N;
    while (lo < hi) {
        int mid = (lo + hi) >> 1;
        if (rows[mid] < v) lo = mid + 1; else hi = mid;
    }
    rp[v] = lo;   // v == V_N naturally yields E_N
}

// ---------------------------------------------------------------------------
// Kernel 2: pack/transpose x (B, Cin, V) -> X0[(b*V + v)*CIN + c]
// 32x32 LDS tile transpose per batch, coalesced on both sides.
// ---------------------------------------------------------------------------
__global__ void cheb_pack_kernel(const float* __restrict__ x, float* __restrict__ X0) {
    __shared__ float tile[32][33];
    const int b  = blockIdx.z;
    const int c0 = blockIdx.y * 32;
    const int v0 = blockIdx.x * 32;
    const int tx = threadIdx.x;   // 0..31
    const int ty = threadIdx.y;   // 0..7
#pragma unroll
    for (int i = 0; i < 4; ++i) {
        int c = c0 + ty + i * 8;
        int v = v0 + tx;
        float val = 0.0f;
        if (v < V_N) val = x[(size_t)b * CIN * V_N + (size_t)c * V_N + v];
        tile[ty + i * 8][tx] = val;
    }
    __syncthreads();
#pragma unroll
    for (int i = 0; i < 4; ++i) {
        int v = v0 + ty + i * 8;
        int c = c0 + tx;
        if (v < V_N) X0[((size_t)b * V_N + v) * CIN + c] = tile[tx][ty + i * 8];
    }
}

// ---------------------------------------------------------------------------
// Kernel 2b: pre-pack weight (K, Cin, Cout) into per-lane WMMA A-fragment
// order so the GEMM loads one aligned float2 (b64) per WMMA:
//   Wt[(((k*16 + kk)*2 + hi)*COUT + m)*2 + j] = W[k][kk*4 + 2*hi + j][m]
// (A 16x4 layout: lane = m, VGPR j holds K = j + 2*hi)
// ---------------------------------------------------------------------------
__global__ void cheb_packw_kernel(const float* __restrict__ W, float* __restrict__ Wt) {
    int idx = blockIdx.x * blockDim.x + threadIdx.x;   // 0 .. KORD*16*2*COUT*2 - 1
    const int total = KORD * 16 * 2 * COUT * 2;
    if (idx >= total) return;
    int j  =  idx        & 1;
    int m  = (idx >> 1)  & (COUT - 1);
    int hi = (idx >> 8)  & 1;
    int kk = (idx >> 9)  & 15;
    int k  =  idx >> 13;
    int cin = kk * 4 + 2 * hi + j;
    Wt[idx] = W[(size_t)k * CIN * COUT + (size_t)cin * COUT + m];
}

// ---------------------------------------------------------------------------
// Kernel 3: SpMM with alpha/beta:  xout = alpha * (L @ xin) + beta * xprev
// Block (64,4): 64 channels per row, 4 rows per block, batch = blockIdx.y.
// Per-thread serial edge loop -> deterministic fp32 accumulation.
// ---------------------------------------------------------------------------
__global__ void cheb_spmm_kernel(const float* __restrict__ xin,
                                 const float* __restrict__ xprev,
                                 float* __restrict__ xout,
                                 const int* __restrict__ rp,
                                 const int* __restrict__ cols,
                                 const float* __restrict__ vals,
                                 float alpha, float beta) {
    const int r = blockIdx.x * 4 + threadIdx.y;
    const int b = blockIdx.y;
    const int c = threadIdx.x;                   // 0..63
    if (r >= V_N) return;
    const int e0 = rp[r], e1 = rp[r + 1];
    float acc = 0.0f;
    for (int e = e0; e < e1; ++e) {
        const int col = cols[e];
        acc += vals[e] * xin[((size_t)b * V_N + col) * CIN + c];
    }
    const size_t o = ((size_t)b * V_N + r) * CIN + c;
    xout[o] = alpha * acc + beta * xprev[o];
}

// ---------------------------------------------------------------------------
// Kernel 4: fused WMMA GEMM:
//   out[b, cout, v] = bias[cout] + sum_k sum_cin W[k][cin][cout] * X_k[b,v,cin]
// A (16x4)  = W_k^T chunk from pre-packed Wt (one b64 load per WMMA)
// B (4x16)  = X_k chunk from LDS (ds_load_2addr_b64, conflict-free stride 68)
// C (16x16) lane dim = N = v  -> coalesced stores into (B, Cout, V).
// Block: 256 threads = 8 waves; wave w owns cout tile m0 = 16*w (full Cout).
// ---------------------------------------------------------------------------
#define LDS_STRIDE 68
__global__ void cheb_gemm_kernel(const float* __restrict__ X,      // 4 buffers, stride xkstride
                                 const float* __restrict__ Wt,     // packed A fragments
                                 const float* __restrict__ bias,   // (Cout)
                                 float* __restrict__ out) {        // (B, Cout, V)
    __shared__ float lds[KORD][16 * LDS_STRIDE];

    const int v0 = blockIdx.x * 16;
    const int b  = blockIdx.y;
    const int t  = threadIdx.x;          // 0..255
    const size_t xkstride = (size_t)B_N * V_N * CIN;

    // Cooperative stage of the 16x64 X_k tiles (float4, coalesced)
    {
        const int vl = t >> 4;           // 0..15 vertex-in-tile
        const int cg = t & 15;           // 0..15 cin group of 4
#pragma unroll
        for (int k = 0; k < KORD; ++k) {
            const float* src = X + k * xkstride +
                               ((size_t)b * V_N + v0 + vl) * CIN + cg * 4;
            float4 d = *reinterpret_cast<const float4*>(src);
            *reinterpret_cast<float4*>(&lds[k][vl * LDS_STRIDE + cg * 4]) = d;
        }
    }
    __syncthreads();

    const int wave = t >> 5;             // 0..7 -> cout tile
    const int lane = t & 31;
    const int nl   = lane & 15;          // N index for B/C, M index for A
    const int hi   = lane >> 4;          // lane half select
    const int m0   = wave * 16;

    // C accumulator initialized with bias: C[r] holds M = m0 + r + 8*hi
    v8f acc;
#pragma unroll
    for (int r = 0; r < 8; ++r) acc[r] = bias[m0 + r + 8 * hi];

#pragma unroll
    for (int k = 0; k < KORD; ++k) {
        const float* Lk = &lds[k][0];
#pragma unroll
        for (int kk = 0; kk < 16; ++kk) {
            const int cin0 = kk * 4 + 2 * hi;      // K = j + 2*hi for VGPR j
            // A fragment: single aligned 8-byte load from packed weights
            v2f a = *reinterpret_cast<const v2f*>(
                &Wt[((((size_t)k * 16 + kk) * 2 + hi) * COUT + m0 + nl) * 2]);
            v2f bf = *reinterpret_cast<const v2f*>(&Lk[nl * LDS_STRIDE + cin0]);
            acc = __builtin_amdgcn_wmma_f32_16x16x4_f32(
                /*neg_a=*/false, a, /*neg_b=*/false, bf,
                /*c_mod=*/(short)0, acc, /*reuse_a=*/false, /*reuse_b=*/false);
        }
    }

    // Store: out[b, m0+r+8*hi, v0+nl] — lanes sweep v -> coalesced 64B segments
#pragma unroll
    for (int r = 0; r < 8; ++r) {
        out[(size_t)b * COUT * V_N + (size_t)(m0 + r + 8 * hi) * V_N + v0 + nl] = acc[r];
    }
}

// ---------------------------------------------------------------------------
// Launcher
// ---------------------------------------------------------------------------
extern "C" void kernel_launch(void* const* d_in, const int* in_sizes, int n_in,
                              void* d_out, int out_size, void* d_ws, size_t ws_size,
                              hipStream_t stream) {
    const float* x      = (const float*)d_in[0];
    const int*   rows   = (const int*)d_in[1];
    const int*   cols   = (const int*)d_in[2];
    const float* vals   = (const float*)d_in[3];
    const float* weight = (const float*)d_in[4];
    const float* bias   = (const float*)d_in[5];
    float*       out    = (float*)d_out;

    char*  ws = (char*)d_ws;
    int*   rp = (int*)ws;                              // (V+1) ints        @ 0
    float* Wt = (float*)(ws + 256 * 1024);             // 128 KB packed W   @ 256K
    float* X  = (float*)(ws + 512 * 1024);             // 4 Chebyshev bufs  @ 512K
    const size_t xk = (size_t)B_N * V_N * CIN;         // elems per buffer

    // 1. CSR row pointers
    cheb_rowptr_kernel<<<(V_N + 1 + 255) / 256, 256, 0, stream>>>(rows, rp);

    // 2. Pack x -> X0 ; pack weights into WMMA A-fragment order
    cheb_pack_kernel<<<dim3((V_N + 31) / 32, CIN / 32, B_N), dim3(32, 8), 0, stream>>>(x, X);
    cheb_packw_kernel<<<(KORD * CIN * COUT + 255) / 256, 256, 0, stream>>>(weight, Wt);

    // 3. Chebyshev recurrence via SpMM
    dim3 sg(V_N / 4, B_N), sb(64, 4);
    // x1 = L x0
    cheb_spmm_kernel<<<sg, sb, 0, stream>>>(X,          X,      X + xk,     rp, cols, vals, 1.0f,  0.0f);
    // x2 = 2 L x1 - x0
    cheb_spmm_kernel<<<sg, sb, 0, stream>>>(X + xk,     X,      X + 2 * xk, rp, cols, vals, 2.0f, -1.0f);
    // x3 = 2 L x2 - x1
    cheb_spmm_kernel<<<sg, sb, 0, stream>>>(X + 2 * xk, X + xk, X + 3 * xk, rp, cols, vals, 2.0f, -1.0f);

    // 4. Fused WMMA GEMM + bias + transpose into (B, Cout, V)
    cheb_gemm_kernel<<<dim3(V_N / 16, B_N), 256, 0, stream>>>(X, Wt, bias, out);
}